// RUMCell_27565100106010
// MI455X (gfx1250) — compile-verified
//
#include <hip/hip_runtime.h>
#include <hip/hip_bf16.h>
#include <math.h>

#define EPSF 1e-12f

typedef float v2f __attribute__((ext_vector_type(2)));
typedef float v8f __attribute__((ext_vector_type(8)));

constexpr int B  = 256;
constexpr int H  = 512;
constexpr int IN = 512;
constexpr int H3 = 1536;

__device__ __forceinline__ float waveReduceSum(float v) {
#pragma unroll
    for (int off = 16; off > 0; off >>= 1)
        v += __shfl_xor(v, off, 32);
    return v;
}

// -------------------------------------------------------------------------
// Kernel 1: proj[m,n] = sum_k inputs[m,k]*W_ih[n,k] + b_ih[n]
//                     + sum_k hidden[m,k]*W_hh[n,k] + b_hh[n]
// One wave per 16x16 output tile, V_WMMA_F32_16X16X4_F32 over K.
// A layout (16x4 f32): lanes 0-15 -> M=0..15, v0=K0,v1=K1; lanes 16-31 -> K2,K3.
// B layout (4x16 f32): lanes 0-15 -> N=0..15, v0=K0,v1=K1; lanes 16-31 -> K2,K3.
// C/D layout: VGPR r -> M = r (+8 for lanes 16-31), N = lane&15.
// -------------------------------------------------------------------------
__global__ __launch_bounds__(128)
void proj_wmma(const float* __restrict__ inputs, const float* __restrict__ hidden,
               const float* __restrict__ W_ih,   const float* __restrict__ b_ih,
               const float* __restrict__ W_hh,   const float* __restrict__ b_hh,
               float* __restrict__ proj) {
    const int lane = threadIdx.x & 31;
    const int wave = threadIdx.x >> 5;
    const int n0   = (blockIdx.x * 4 + wave) * 16;   // proj column tile (0..1535)
    const int m0   = blockIdx.y * 16;                // batch row tile
    const int half = lane >> 4;                      // 0: K={0,1}, 1: K={2,3}
    const int l    = lane & 15;

    const float* aih = inputs + (size_t)(m0 + l) * IN + half * 2;
    const float* ahh = hidden + (size_t)(m0 + l) * IN + half * 2;
    const float* bih = W_ih   + (size_t)(n0 + l) * IN + half * 2;  // B[k][n] = W[n*IN+k]
    const float* bhh = W_hh   + (size_t)(n0 + l) * IN + half * 2;

    v8f acc = {};
#pragma unroll 4
    for (int k = 0; k < IN; k += 4) {
        v2f a0 = *(const v2f*)(aih + k);
        v2f b0 = *(const v2f*)(bih + k);
        acc = __builtin_amdgcn_wmma_f32_16x16x4_f32(false, a0, false, b0,
                                                    (short)0, acc, false, false);
        v2f a1 = *(const v2f*)(ahh + k);
        v2f b1 = *(const v2f*)(bhh + k);
        acc = __builtin_amdgcn_wmma_f32_16x16x4_f32(false, a1, false, b1,
                                                    (short)0, acc, false, false);
    }

    const int   n    = n0 + l;
    const float bias = b_ih[n] + b_hh[n];
#pragma unroll
    for (int r = 0; r < 8; ++r) {
        const int m = m0 + r + half * 8;
        proj[(size_t)m * H3 + n] = acc[r] + bias;
    }
}

// -------------------------------------------------------------------------
// Kernel 2: per batch row, compute rh = R @ hidden analytically.
//   u = x/||x||, v = normalize(r - (u.r)u), cos = u.r_n, sin = sqrt(1-cos^2)
//   R h = h + (cos-1)(u(u.h)+v(v.h)) + sin(v(u.h)-u(v.h))
// Only needs 5 scalar reductions per row; never materializes R.
// -------------------------------------------------------------------------
__global__ __launch_bounds__(256)
void rot_kernel(const float* __restrict__ proj, const float* __restrict__ hidden,
                float* __restrict__ rh_out) {
    const int b = blockIdx.x;
    const float* row = proj + (size_t)b * H3;
    const float* rg  = row + H;            // r
    const float* xg  = row + 2 * H;        // x_emb
    const float* hg  = hidden + (size_t)b * H;

    float xx = 0.f, rr = 0.f, xr = 0.f, xh = 0.f, rhd = 0.f;
    for (int i = threadIdx.x; i < H; i += 256) {
        float xi = xg[i], ri = rg[i], hi = hg[i];
        xx = fmaf(xi, xi, xx);
        rr = fmaf(ri, ri, rr);
        xr = fmaf(xi, ri, xr);
        xh = fmaf(xi, hi, xh);
        rhd = fmaf(ri, hi, rhd);
    }

    __shared__ float s[5][8];
    const int wave = threadIdx.x >> 5, lane = threadIdx.x & 31;
    xx = waveReduceSum(xx); rr = waveReduceSum(rr); xr = waveReduceSum(xr);
    xh = waveReduceSum(xh); rhd = waveReduceSum(rhd);
    if (lane == 0) {
        s[0][wave] = xx; s[1][wave] = rr; s[2][wave] = xr;
        s[3][wave] = xh; s[4][wave] = rhd;
    }
    __syncthreads();
    xx = rr = xr = xh = rhd = 0.f;
#pragma unroll
    for (int w = 0; w < 8; ++w) {
        xx += s[0][w]; rr += s[1][w]; xr += s[2][w];
        xh += s[3][w]; rhd += s[4][w];
    }

    const float inx   = 1.f / fmaxf(sqrtf(xx), EPSF);
    const float inr   = 1.f / fmaxf(sqrtf(rr), EPSF);
    const float costh = xr * inx * inr;
    const float sinth = sqrtf(fmaxf(1.f - costh * costh, 0.f));
    const float ur    = xr * inx;                               // u . r
    const float inw   = 1.f / fmaxf(sqrtf(fmaxf(rr - ur * ur, 0.f)), EPSF);
    const float uh    = xh * inx;                               // u . h
    const float vh    = (rhd - ur * uh) * inw;                  // v . h
    const float cm1   = costh - 1.f;

    for (int i = threadIdx.x; i < H; i += 256) {
        float ui = xg[i] * inx;
        float vi = (rg[i] - ur * ui) * inw;
        rh_out[(size_t)b * H + i] =
            hg[i] + cm1 * (ui * uh + vi * vh) + sinth * (vi * uh - ui * vh);
    }
}

// -------------------------------------------------------------------------
// Kernel 3: hidden_new = assoc_mem[b] @ rh[b]  (the 268 MB HBM-bound pass),
// fused with: c = relu(hidden_new + x_emb); t = ug*h + (1-ug)*c; out = l2norm(t).
// One block (8 waves) per batch row; each wave streams 64 matrix rows with
// coalesced float4 loads and reduces with wave32 shuffles.
// -------------------------------------------------------------------------
__global__ __launch_bounds__(256)
void matvec_norm_kernel(const float* __restrict__ assoc, const float* __restrict__ proj,
                        const float* __restrict__ hidden, const float* __restrict__ rh,
                        float* __restrict__ out) {
    const int b = blockIdx.x;
    __shared__ float s_rh[H];
    __shared__ float s_tmp[H];
    __shared__ float s_red[8];

    for (int i = threadIdx.x; i < H; i += 256)
        s_rh[i] = rh[(size_t)b * H + i];
    __syncthreads();

    const int wave = threadIdx.x >> 5, lane = threadIdx.x & 31;
    const float* A   = assoc + (size_t)b * H * H;
    const float* row = proj + (size_t)b * H3;

    for (int i = wave * 64; i < wave * 64 + 64; ++i) {
        const float4* Ar = (const float4*)(A + (size_t)i * H);
        if (i + 1 < H)  // hint the next 2KB row into cache (global_prefetch_b8)
            __builtin_prefetch(A + (size_t)(i + 1) * H + lane * 16, 0, 0);
        float acc = 0.f;
#pragma unroll
        for (int s4 = 0; s4 < 4; ++s4) {
            float4 a4 = Ar[s4 * 32 + lane];       // wave reads 512B contiguous
            const int j = (s4 * 32 + lane) * 4;
            acc = fmaf(a4.x, s_rh[j + 0], acc);
            acc = fmaf(a4.y, s_rh[j + 1], acc);
            acc = fmaf(a4.z, s_rh[j + 2], acc);
            acc = fmaf(a4.w, s_rh[j + 3], acc);
        }
        acc = waveReduceSum(acc);
        if (lane == 0) {
            float ug = row[i];                    // raw u_gate (no sigmoid in ref)
            float c  = fmaxf(acc + row[2 * H + i], 0.f);
            s_tmp[i] = ug * hidden[(size_t)b * H + i] + (1.f - ug) * c;
        }
    }
    __syncthreads();

    float ss = 0.f;
    for (int i = threadIdx.x; i < H; i += 256) {
        float t = s_tmp[i];
        ss = fmaf(t, t, ss);
    }
    ss = waveReduceSum(ss);
    if (lane == 0) s_red[wave] = ss;
    __syncthreads();
    float tot = 0.f;
#pragma unroll
    for (int w = 0; w < 8; ++w) tot += s_red[w];
    const float scale = 1.f / fmaxf(sqrtf(tot), EPSF);   // ETA == 1.0

    for (int i = threadIdx.x; i < H; i += 256)
        out[(size_t)b * H + i] = s_tmp[i] * scale;
}

// -------------------------------------------------------------------------
extern "C" void kernel_launch(void* const* d_in, const int* in_sizes, int n_in,
                              void* d_out, int out_size, void* d_ws, size_t ws_size,
                              hipStream_t stream) {
    const float* inputs = (const float*)d_in[0];
    const float* hidden = (const float*)d_in[1];
    const float* assoc  = (const float*)d_in[2];
    const float* W_ih   = (const float*)d_in[3];
    const float* b_ih   = (const float*)d_in[4];
    const float* W_hh   = (const float*)d_in[5];
    const float* b_hh   = (const float*)d_in[6];
    float* out = (float*)d_out;

    float* proj = (float*)d_ws;                    // B*H3 floats = 1.5 MB
    float* rh   = proj + (size_t)B * H3;           // B*H  floats = 0.5 MB

    // 96 column tiles x 16 row tiles; 4 waves (4 n-tiles) per block.
    proj_wmma<<<dim3(H3 / 64, B / 16), 128, 0, stream>>>(
        inputs, hidden, W_ih, b_ih, W_hh, b_hh, proj);

    rot_kernel<<<B, 256, 0, stream>>>(proj, hidden, rh);

    matvec_norm_kernel<<<B, 256, 0, stream>>>(assoc, proj, hidden, rh, out);
}